// PDEBoundaryOperators2D_52536039965344
// MI455X (gfx1250) — compile-verified
//
#include <hip/hip_runtime.h>
#include <hip/hip_bf16.h>

// CDNA5 (gfx1250) wave32 WMMA types
typedef __attribute__((ext_vector_type(2))) float v2f;  // A/B for V_WMMA_F32_16X16X4_F32
typedef __attribute__((ext_vector_type(8))) float v8f;  // 16x16 f32 C/D

// One wave computes one 16x16 tile of the pairwise operator matrix.
// s_ij = ||x1_i||^2 - 2<x1_i, x2_j> + ||x2_j||^2 is produced by a single
// v_wmma_f32_16x16x4_f32 with:
//   A row i  = [x, y, z, 1]                      (16x4)
//   B col j  = [-2x', -2y', -2z', ||x2_j||^2]    (4x16)
//   C[i][j]  = ||x1_i||^2                        (accumulator pre-load)
// Epilogue: out = v^2 * exp(-s/(2 l^2)) * (s^2 - 10 l^2 s + 15 l^4) / l^8
__global__ __launch_bounds__(256)
void pde_biharm_rbf_wmma(const float* __restrict__ X1,
                         const float* __restrict__ X2,
                         const float* __restrict__ lsp,
                         const float* __restrict__ varp,
                         float* __restrict__ out,
                         int n1, int n2) {
    const int tid  = threadIdx.x;
    const int wave = tid >> 5;          // 8 waves per block, tiled along N
    const int lane = tid & 31;
    const int half = lane >> 4;         // 0: lanes 0-15, 1: lanes 16-31
    const int lm   = lane & 15;

    const int j0 = (blockIdx.x * 8 + wave) * 16;  // column tile origin
    const int i0 = blockIdx.y * 16;               // row tile origin
    if (j0 >= n2 || i0 >= n1) return;             // wave-uniform guard

    // ---- per-lane row loads (broadcast across the 2 lanes sharing lm; L2 resident)
    const float* x1r = X1 + (size_t)(i0 + lm) * 3;
    const float a0 = x1r[0], a1 = x1r[1], a2 = x1r[2];
    const float* x2r = X2 + (size_t)(j0 + lm) * 3;
    const float b0 = x2r[0], b1 = x2r[1], b2 = x2r[2];

    const float norm1 = a0 * a0 + a1 * a1 + a2 * a2;   // ||x1_{i0+lm}||^2
    const float norm2 = b0 * b0 + b1 * b1 + b2 * b2;   // ||x2_{j0+lm}||^2

    // ---- A (16x4, M=lm): lanes 0-15 hold K={0,1}, lanes 16-31 hold K={2,3}
    v2f A;
    A.x = half ? a2 : a0;
    A.y = half ? 1.0f : a1;

    // ---- B (4x16, N=lm): same K split across lane halves / VGPRs
    v2f B;
    B.x = half ? (-2.0f * b2) : (-2.0f * b0);
    B.y = half ? norm2 : (-2.0f * b1);

    // ---- C pre-load: C VGPR r holds row M = r + 8*half -> needs norm1 from lane r+8*half
    v8f C;
#pragma unroll
    for (int r = 0; r < 8; ++r)
        C[r] = __shfl(norm1, r + 8 * half, 32);   // ds_bpermute_b32, wave32

    // ---- D = A*B + C = pairwise squared distance tile
    v8f S = __builtin_amdgcn_wmma_f32_16x16x4_f32(
        /*neg_a=*/false, A, /*neg_b=*/false, B,
        /*c_mod=*/(short)0, C, /*reuse_a=*/false, /*reuse_b=*/false);

    // ---- scalar constants (D = 3): 2D+4 = 10, D(D+2) = 15
    const float ls  = lsp[0];
    const float var = varp[0];
    const float l2      = ls * ls;
    const float inv_l2  = 1.0f / l2;
    const float nh      = -0.5f * inv_l2;                      // exp argument scale
    const float c1      = 10.0f * l2;
    const float c2      = 15.0f * l2 * l2;
    const float inv_l8  = (inv_l2 * inv_l2) * (inv_l2 * inv_l2);
    const float scale   = (var * var) * inv_l8;

    // ---- epilogue + store: lane (half,lm) writes rows M = r + 8*half, col j0+lm.
    // Each global_store_b32 covers two contiguous 64B row segments per wave.
    float* ocol = out + (size_t)i0 * n2 + (size_t)j0 + lm;
#pragma unroll
    for (int r = 0; r < 8; ++r) {
        const float s    = S[r];
        const float poly = s * s - c1 * s + c2;
        const float val  = scale * __expf(nh * s) * poly;
        ocol[(size_t)(r + 8 * half) * n2] = val;
    }
}

extern "C" void kernel_launch(void* const* d_in, const int* in_sizes, int n_in,
                              void* d_out, int out_size, void* d_ws, size_t ws_size,
                              hipStream_t stream) {
    const float* X1  = (const float*)d_in[0];   // [n1, 3]
    const float* X2  = (const float*)d_in[1];   // [n2, 3]
    const float* ls  = (const float*)d_in[2];   // scalar
    const float* var = (const float*)d_in[3];   // scalar
    float* out = (float*)d_out;                 // [n1, n2]

    const int n1 = in_sizes[0] / 3;             // 4096
    const int n2 = in_sizes[1] / 3;             // 4096

    const int tiles_n = (n2 + 15) / 16;         // 16-wide column tiles
    const int tiles_m = (n1 + 15) / 16;         // 16-tall row tiles
    dim3 grid((tiles_n + 7) / 8, tiles_m);      // 8 waves (tiles) per block along N
    dim3 block(256);
    pde_biharm_rbf_wmma<<<grid, block, 0, stream>>>(X1, X2, ls, var, out, n1, n2);
}